// FpQuantizer_52742198395451
// MI455X (gfx1250) — compile-verified
//
#include <hip/hip_runtime.h>

// ---------------------------------------------------------------------------
// FP quantize to e4m3-style format (EXP=4, MAN=3, bias 7, subnormal floor
// 2^-6 -> grid 2^-9, saturate at +/-240), output back as f32.
//
// MI455X roofline: 134.2M f32 in + out = ~1.07 GB @ 23.3 TB/s => ~46us floor.
// Pure streaming kernel:
//   * clamp to [-240, 240], then hardware E4M3 round-trip
//     (v_cvt_pk_fp8_f32 -> v_cvt_pk_f32_fp8): identical grid / RNE / denorm
//     behavior to the reference math for all inputs once pre-clamped.
//   * one 128-bit non-temporal load + store per thread, no tail branches in
//     the hot kernel (remainder handled by a separate scalar kernel).
// ---------------------------------------------------------------------------

typedef __attribute__((ext_vector_type(4))) float v4f;
typedef __attribute__((ext_vector_type(2))) float v2f;

#if __has_builtin(__builtin_amdgcn_cvt_pk_fp8_f32) && \
    __has_builtin(__builtin_amdgcn_cvt_pk_f32_fp8)
#define USE_HW_FP8 1
#else
#define USE_HW_FP8 0
#endif

__device__ __forceinline__ float clamp240(float x) {
    return fminf(fmaxf(x, -240.0f), 240.0f);
}

#if USE_HW_FP8

__device__ __forceinline__ v4f quant4(v4f v) {
    float a = clamp240(v.x);
    float b = clamp240(v.y);
    float c = clamp240(v.z);
    float d = clamp240(v.w);
    // 4 floats -> 4 fp8 E4M3 (RNE, denorms preserved) packed in one dword
    int p = __builtin_amdgcn_cvt_pk_fp8_f32(a, b, 0, false); // low word
    p = __builtin_amdgcn_cvt_pk_fp8_f32(c, d, p, true);      // high word
    // exact unpack back to f32
    v2f lo = __builtin_amdgcn_cvt_pk_f32_fp8(p, false);
    v2f hi = __builtin_amdgcn_cvt_pk_f32_fp8(p, true);
    v4f r;
    r.x = lo.x; r.y = lo.y; r.z = hi.x; r.w = hi.y;
    return r;
}

__device__ __forceinline__ float quant1(float x) {
    float c = clamp240(x);
    int p = __builtin_amdgcn_cvt_pk_fp8_f32(c, c, 0, false);
    v2f lo = __builtin_amdgcn_cvt_pk_f32_fp8(p, false);
    return lo.x;
}

#else  // software RNE fallback (branchless), same grid as reference

__device__ __forceinline__ float quant1(float x) {
    float ax = fminf(fabsf(x), 240.0f);
    // normals: RNE-truncate mantissa to 3 bits via integer rounding
    unsigned u = __float_as_uint(ax);
    unsigned lsb = (u >> 20) & 1u;
    u += 0x7FFFFu + lsb;
    u &= 0xFFF00000u;
    float qn = fminf(__uint_as_float(u), 240.0f);
    // subnormals (|x| < 2^-6): RNE onto fixed 2^-9 grid via magic add
    float qs = (ax + 16384.0f) - 16384.0f;   // 2^14 has ulp 2^-9
    float q = (ax < 0.015625f) ? qs : qn;
    return (x < 0.0f) ? -q : q;
}

__device__ __forceinline__ v4f quant4(v4f v) {
    v4f r;
    r.x = quant1(v.x); r.y = quant1(v.y);
    r.z = quant1(v.z); r.w = quant1(v.w);
    return r;
}

#endif

// Hot kernel: exactly one b128 NT load + one b128 NT store per thread.
__global__ void __launch_bounds__(256)
fpquant_vec_kernel(const float* __restrict__ x, float* __restrict__ out,
                   long long n4) {
    long long t = (long long)blockIdx.x * 256 + threadIdx.x;  // vector index
    if (t >= n4) return;
    long long i = t << 2;
    v4f v = __builtin_nontemporal_load((const v4f*)(x + i));
    v4f r = quant4(v);
    __builtin_nontemporal_store(r, (v4f*)(out + i));
}

// Cold kernel: scalar remainder (launched only when n % 4 != 0).
__global__ void fpquant_tail_kernel(const float* __restrict__ x,
                                    float* __restrict__ out,
                                    long long start, long long n) {
    long long j = start + threadIdx.x;
    if (j < n) out[j] = quant1(x[j]);
}

extern "C" void kernel_launch(void* const* d_in, const int* in_sizes, int n_in,
                              void* d_out, int out_size, void* d_ws, size_t ws_size,
                              hipStream_t stream) {
    const float* x = (const float*)d_in[0];
    float* out = (float*)d_out;
    long long n = (long long)out_size;

    long long n4 = n >> 2;            // full 4-element vectors
    long long rem = n - (n4 << 2);

    if (n4 > 0) {
        const int threads = 256;      // 8 wave32s per block
        unsigned blocks = (unsigned)((n4 + threads - 1) / threads);
        fpquant_vec_kernel<<<blocks, threads, 0, stream>>>(x, out, n4);
    }
    if (rem > 0) {
        fpquant_tail_kernel<<<1, 32, 0, stream>>>(x, out, n4 << 2, n);
    }
}